// GNN_25391846654579
// MI455X (gfx1250) — compile-verified
//
#include <hip/hip_runtime.h>

#define N_NODES 100000
#define DIM 64
#define LDS_STRIDE 68   // pad 64->68 floats so B reads hit distinct LDS banks

typedef __attribute__((ext_vector_type(2))) float v2f;
typedef __attribute__((ext_vector_type(8))) float v8f;

// ---------------------------------------------------------------------------
// Degree count: one thread per edge, atomic add 1.0 to cnt[dst]
// ---------------------------------------------------------------------------
__global__ void sage_count(const int* __restrict__ edge, float* __restrict__ cnt, int E) {
    int e = blockIdx.x * blockDim.x + threadIdx.x;
    if (e < E) {
        int dst = edge[E + e];                 // row 1 of edge_index
        atomicAdd(&cnt[dst], 1.0f);
    }
}

// ---------------------------------------------------------------------------
// Feature scatter: 64 threads per edge, s[dst,f] += x[src,f]
// Coalesced reads of x[src] row, f32 atomic adds into s.
// ---------------------------------------------------------------------------
__global__ void sage_scatter(const float* __restrict__ x, const int* __restrict__ edge,
                             float* __restrict__ s, int E) {
    long long tid = (long long)blockIdx.x * blockDim.x + threadIdx.x;
    int e = (int)(tid >> 6);
    int f = (int)(tid & 63);
    if (e < E) {
        int src = edge[e];
        int dst = edge[E + e];
        atomicAdd(&s[(long long)dst * DIM + f], x[(long long)src * DIM + f]);
    }
}

// ---------------------------------------------------------------------------
// Fused SAGE layer GEMM:
//   out = relu( (s / max(cnt,1)) @ Wl^T + bl + x @ Wr^T )
// One wave32 per 16-row node block; V_WMMA_F32_16X16X4_F32, 16 K-steps,
// 4 column tiles (16x64 output per wave). Weights staged in LDS.
// ---------------------------------------------------------------------------
__global__ __launch_bounds__(256)
void sage_gemm(const float* __restrict__ s, const float* __restrict__ cnt,
               const float* __restrict__ x,
               const float* __restrict__ Wl, const float* __restrict__ bl,
               const float* __restrict__ Wr, float* __restrict__ out) {
    __shared__ float lwl[DIM * LDS_STRIDE];
    __shared__ float lwr[DIM * LDS_STRIDE];
    __shared__ float lb[DIM];

    const int t256 = threadIdx.y * 32 + threadIdx.x;
    // Cooperative weight staging: 4096 elements each, 16 per thread
    for (int i = t256; i < DIM * DIM; i += 256) {
        int n = i >> 6, k = i & 63;
        lwl[n * LDS_STRIDE + k] = Wl[i];
        lwr[n * LDS_STRIDE + k] = Wr[i];
    }
    if (t256 < DIM) lb[t256] = bl[t256];
    __syncthreads();

    const int rowBlock = blockIdx.x * blockDim.y + threadIdx.y;
    const int rowBase  = rowBlock * 16;
    if (rowBase >= N_NODES) return;            // wave-uniform: EXEC stays all-ones

    const int lane = threadIdx.x;              // 0..31
    const int m    = lane & 15;                // A row / D column within tile
    const int g    = lane >> 4;                // half-wave selector
    const int mrow = rowBase + m;

    const float inv = 1.0f / fmaxf(cnt[mrow], 1.0f);

    // A-operand layout for 16x4 f32: lane holds A[m][2g + {0,1}] per K-chunk
    const float* srow = s + (long long)mrow * DIM + 2 * g;
    const float* xrow = x + (long long)mrow * DIM + 2 * g;

    v8f acc[4];
    acc[0] = v8f{}; acc[1] = v8f{}; acc[2] = v8f{}; acc[3] = v8f{};

    #pragma unroll
    for (int t = 0; t < 16; ++t) {             // K = 64 in chunks of 4
        const int k0 = 4 * t + 2 * g;
        v2f a_agg = *(const v2f*)(srow + 4 * t);
        a_agg *= inv;                          // fold mean division into A
        const v2f a_x = *(const v2f*)(xrow + 4 * t);

        #pragma unroll
        for (int c = 0; c < 4; ++c) {          // 4 column tiles of 16
            const int n = c * 16 + m;
            const v2f bL = *(const v2f*)&lwl[n * LDS_STRIDE + k0];
            const v2f bR = *(const v2f*)&lwr[n * LDS_STRIDE + k0];
            acc[c] = __builtin_amdgcn_wmma_f32_16x16x4_f32(
                         false, a_agg, false, bL, (short)0, acc[c], false, false);
            acc[c] = __builtin_amdgcn_wmma_f32_16x16x4_f32(
                         false, a_x,   false, bR, (short)0, acc[c], false, false);
        }
    }

    // D layout: VGPR r -> M = r (lanes 0-15) or r+8 (lanes 16-31); N = m
    #pragma unroll
    for (int c = 0; c < 4; ++c) {
        const int n = c * 16 + m;
        const float bias = lb[n];
        #pragma unroll
        for (int r = 0; r < 8; ++r) {
            const int mo = rowBase + r + 8 * g;
            const float v = acc[c][r] + bias;
            out[(long long)mo * DIM + n] = fmaxf(v, 0.0f);
        }
    }
}

// ---------------------------------------------------------------------------
extern "C" void kernel_launch(void* const* d_in, const int* in_sizes, int n_in,
                              void* d_out, int out_size, void* d_ws, size_t ws_size,
                              hipStream_t stream) {
    const float* x    = (const float*)d_in[0];
    const int*   edge = (const int*)d_in[1];   // int32 (JAX x64-disabled canonicalization)
    const float* W1l  = (const float*)d_in[2];
    const float* b1l  = (const float*)d_in[3];
    const float* W1r  = (const float*)d_in[4];
    const float* W2l  = (const float*)d_in[5];
    const float* b2l  = (const float*)d_in[6];
    const float* W2r  = (const float*)d_in[7];
    float* out = (float*)d_out;

    const int E = in_sizes[1] / 2;             // 1,600,000

    // Workspace layout: [cnt: N f32][s: N*64 f32][h1: N*64 f32]  (~51.6 MB)
    char* ws = (char*)d_ws;
    size_t cntBytes = ((size_t)N_NODES * 4 + 255) & ~(size_t)255;
    size_t matBytes = (size_t)N_NODES * DIM * 4;
    float* cnt = (float*)ws;
    float* s   = (float*)(ws + cntBytes);
    float* h1  = (float*)(ws + cntBytes + matBytes);

    hipMemsetAsync(cnt, 0, (size_t)N_NODES * 4, stream);
    hipMemsetAsync(s,   0, matBytes, stream);

    // Degree counts (shared by both layers)
    sage_count<<<(E + 255) / 256, 256, 0, stream>>>(edge, cnt, E);

    const long long scatterThreads = (long long)E * DIM;
    const int scatterBlocks = (int)((scatterThreads + 255) / 256);
    const int rowBlocks  = (N_NODES + 15) / 16;        // 6250
    const int gemmBlocks = (rowBlocks + 7) / 8;        // 8 waves per workgroup
    const dim3 gemmBlockDim(32, 8);

    // Layer 1
    sage_scatter<<<scatterBlocks, 256, 0, stream>>>(x, edge, s, E);
    sage_gemm<<<gemmBlocks, gemmBlockDim, 0, stream>>>(s, cnt, x, W1l, b1l, W1r, h1);

    // Layer 2
    hipMemsetAsync(s, 0, matBytes, stream);
    sage_scatter<<<scatterBlocks, 256, 0, stream>>>(h1, edge, s, E);
    sage_gemm<<<gemmBlocks, gemmBlockDim, 0, stream>>>(s, cnt, h1, W2l, b2l, W2r, out);
}